// Projector_58574763983326
// MI455X (gfx1250) — compile-verified
//
#include <hip/hip_runtime.h>
#include <hip/hip_bf16.h>
#include <math.h>

// Problem constants (from the reference)
#define NB   128     // batch
#define DD   64      // bond dimension
#define NIN  64
#define NOUT 65
#define NMID 63      // scan length
#define S68  68      // LDS row stride (DWORDs) -> conflict-free b64 fragment loads
#define MATF (DD * S68)   // floats per 64x64 LDS matrix (4352)

typedef float v2f __attribute__((ext_vector_type(2)));
typedef float v8f __attribute__((ext_vector_type(8)));

// ---------------------------------------------------------------------------
// 64x64x64 f32 matmul on LDS using V_WMMA_F32_16X16X4_F32.
//   C[m][n] (+)= sum_k Aop[m][k] * B[k][n]
//   TRANSA=false: Aop[m][k] = A[m*S68+k] ; TRANSA=true: Aop[m][k] = A[k*S68+m]
// 256 threads = 8 wave32; wave w computes output tiles 2w and 2w+1 of the
// 4x4 grid of 16x16 tiles (both share tm -> compiler CSEs the A fragment).
// EXEC is all-ones at every call site (WMMA requirement).
// ---------------------------------------------------------------------------
template <bool TRANSA>
__device__ __forceinline__ void mm64(const float* __restrict__ A,
                                     const float* __restrict__ B,
                                     float* __restrict__ C,
                                     int tid, bool accum) {
  const int lane = tid & 31;
  const int w    = tid >> 5;
  const int half = lane >> 4;   // 0: lanes 0-15, 1: lanes 16-31
  const int l15  = lane & 15;
  #pragma unroll
  for (int tt = 0; tt < 2; ++tt) {
    const int t  = w * 2 + tt;
    const int tm = t >> 2, tn = t & 3;
    const int mrow = tm * 16 + l15;
    const int ncol = tn * 16 + l15;
    v8f acc;
    if (accum) {
      #pragma unroll
      for (int v = 0; v < 8; ++v)
        acc[v] = C[(tm * 16 + v + half * 8) * S68 + ncol];
    } else {
      #pragma unroll
      for (int v = 0; v < 8; ++v) acc[v] = 0.0f;
    }
    #pragma unroll
    for (int kk = 0; kk < 16; ++kk) {
      const int kb = kk * 4 + half * 2;   // K pair handled by this lane-half
      v2f a, bf;
      if (TRANSA) {
        a.x = A[kb * S68 + mrow];
        a.y = A[(kb + 1) * S68 + mrow];
      } else {
        a.x = A[mrow * S68 + kb];
        a.y = A[mrow * S68 + kb + 1];
      }
      bf.x = B[kb * S68 + ncol];
      bf.y = B[(kb + 1) * S68 + ncol];
      acc = __builtin_amdgcn_wmma_f32_16x16x4_f32(false, a, false, bf,
                                                  (short)0, acc, false, false);
    }
    #pragma unroll
    for (int v = 0; v < 8; ++v)
      C[(tm * 16 + v + half * 8) * S68 + ncol] = acc[v];
  }
}

// Wave32 butterfly + 8-slot LDS exchange: 2 barriers instead of 9.
__device__ __forceinline__ float block_reduce_max(float v, float* red, int tid) {
  #pragma unroll
  for (int off = 16; off > 0; off >>= 1)
    v = fmaxf(v, __shfl_xor(v, off, 32));
  if ((tid & 31) == 0) red[tid >> 5] = v;
  __syncthreads();
  float r = red[0];
  #pragma unroll
  for (int i = 1; i < 8; ++i) r = fmaxf(r, red[i]);
  __syncthreads();   // protect red[] for reuse
  return r;
}

__device__ __forceinline__ float block_reduce_sum(float v, float* red, int tid) {
  #pragma unroll
  for (int off = 16; off > 0; off >>= 1)
    v += __shfl_xor(v, off, 32);
  if ((tid & 31) == 0) red[tid >> 5] = v;
  __syncthreads();
  float r = red[0];
  #pragma unroll
  for (int i = 1; i < 8; ++i) r += red[i];
  __syncthreads();
  return r;
}

// Prefetch `bytes` starting at p, one 128B line per slot, strided over threads.
__device__ __forceinline__ void prefetch_block(const char* p, int bytes, int tid) {
  for (int off = tid * 128; off < bytes; off += 256 * 128)
    __builtin_prefetch(p + off, 0, 3);   // -> global_prefetch_b8
}

// ---------------------------------------------------------------------------
// Kernel 1: one workgroup per batch. Fully fused MPS scan in LDS.
// x:  (129,128,2)   Mi: (64,64,64,2)   Mo: (65,64,64,2,2)   out: (129,)
// ---------------------------------------------------------------------------
__global__ void __launch_bounds__(256)
batch_chain_kernel(const float* __restrict__ x,
                   const float* __restrict__ Mi,
                   const float* __restrict__ Mo,
                   float* __restrict__ out) {
  extern __shared__ float smem[];
  float* P0  = smem;             // Al ping
  float* P1  = P0 + MATF;        // Al pong
  float* Ain = P1 + MATF;
  float* M0b = Ain + MATF;
  float* M1b = M0b + MATF;
  float* T0b = M1b + MATF;       // also holds X0 transiently
  float* T1b = T0b + MATF;       // also holds X1 transiently
  float* red = T1b + MATF;       // 256 (only 8 slots used by reductions)
  float* vb  = red + 256;        // 128  (A0 / Xl)
  float* vb2 = vb + 128;         // 128  (v_j)

  const int tid = threadIdx.x;
  const int b   = blockIdx.x;

  // --- A0[u][j] = x[0,b,:] . Mo[0, l=0, u, :, j];  Al = A0 A0^T -------------
  if (tid < 128) {
    const int u = tid >> 1, j = tid & 1;
    const float x0 = x[b * 2 + 0];
    const float x1 = x[b * 2 + 1];
    const float* m4 = Mo + (size_t)u * 4;        // Mo[0,0,u,:,:]
    vb[tid] = x0 * m4[j] + x1 * m4[2 + j];
  }
  // warm the first step's parameter slabs while we build Al
  prefetch_block((const char*)Mi, 4096 * 2 * 4, tid);
  prefetch_block((const char*)(Mo + (size_t)4096 * 4), 4096 * 4 * 4, tid);
  __syncthreads();
  for (int idx = tid; idx < 4096; idx += 256) {
    const int u = idx >> 6, d = idx & 63;
    P0[u * S68 + d] = vb[u * 2] * vb[d * 2] + vb[u * 2 + 1] * vb[d * 2 + 1];
  }
  __syncthreads();

  float logsum = 0.0f;
  float* Al  = P0;
  float* Aln = P1;

  // --- 63 scan steps --------------------------------------------------------
  for (int k = 0; k < NMID; ++k) {
    const float xin0 = x[((2 * k + 1) * NB + b) * 2 + 0];
    const float xin1 = x[((2 * k + 1) * NB + b) * 2 + 1];
    const float xo0  = x[((2 * k + 2) * NB + b) * 2 + 0];
    const float xo1  = x[((2 * k + 2) * NB + b) * 2 + 1];
    const float* mi = Mi + (size_t)k * 4096 * 2;
    const float* mo = Mo + (size_t)(k + 1) * 4096 * 4;

    // build A_in, X0, X1 in LDS; fold in local |Al| max scan
    float lmax = 0.0f;
    for (int idx = tid; idx < 4096; idx += 256) {
      const int l = idx >> 6, r = idx & 63;
      Ain[l * S68 + r] = xin0 * mi[idx * 2] + xin1 * mi[idx * 2 + 1];
      const float* m4 = mo + (size_t)idx * 4;   // Mo[k+1, l, r, :, :]
      T0b[l * S68 + r] = xo0 * m4[0] + xo1 * m4[2];   // X0
      T1b[l * S68 + r] = xo0 * m4[1] + xo1 * m4[3];   // X1
      lmax = fmaxf(lmax, fabsf(Al[l * S68 + r]));
    }
    // overlap next step's (and tail's) parameter fetch with the 6 matmuls:
    // Mi[k+1] valid for k<=62 (tail uses Mi[63]); Mo[k+2] valid for k<=62.
    prefetch_block((const char*)(mi + 4096 * 2), 4096 * 2 * 4, tid);
    prefetch_block((const char*)(mo + 4096 * 4), 4096 * 4 * 4, tid);

    const float amax = block_reduce_max(lmax, red, tid);
    logsum += logf(amax);
    const float inv = 1.0f / amax;
    for (int idx = tid; idx < 4096; idx += 256) {
      const int l = idx >> 6, r = idx & 63;
      Al[l * S68 + r] *= inv;
    }
    __syncthreads();

    // M_i = A_in x X_i
    mm64<false>(Ain, T0b, M0b, tid, false);
    mm64<false>(Ain, T1b, M1b, tid, false);
    __syncthreads();
    // T_i = M_i^T x Aln   (overwrites X_i, no longer needed)
    mm64<true>(M0b, Al, T0b, tid, false);
    mm64<true>(M1b, Al, T1b, tid, false);
    __syncthreads();
    // Al' = T0 x M0 + T1 x M1
    mm64<false>(T0b, M0b, Aln, tid, false);
    mm64<false>(T1b, M1b, Aln, tid, true);
    __syncthreads();
    float* t = Al; Al = Aln; Aln = t;
  }

  // --- last (rank-1 column) step -------------------------------------------
  {
    const float xin0 = x[(127 * NB + b) * 2 + 0];
    const float xin1 = x[(127 * NB + b) * 2 + 1];
    const float xo0  = x[(128 * NB + b) * 2 + 0];
    const float xo1  = x[(128 * NB + b) * 2 + 1];
    const float* mi = Mi + (size_t)63 * 4096 * 2;
    const float* mo = Mo + (size_t)64 * 4096 * 4;

    float lmax = 0.0f;
    for (int idx = tid; idx < 4096; idx += 256) {
      const int l = idx >> 6, r = idx & 63;
      Ain[l * S68 + r] = xin0 * mi[idx * 2] + xin1 * mi[idx * 2 + 1];
      lmax = fmaxf(lmax, fabsf(Al[l * S68 + r]));
    }
    if (tid < 128) {             // Xl[r][j] = x . Mo[64, r, 0, :, j]
      const int r = tid >> 1, j = tid & 1;
      const float* m4 = mo + (size_t)(r * 64) * 4;
      vb[tid] = xo0 * m4[j] + xo1 * m4[2 + j];
    }
    __syncthreads();             // Ain / vb visible
    const float amax = block_reduce_max(lmax, red, tid);
    // v_j[l] = sum_r Ain[l][r] * Xl[r][j]
    if (tid < 128) {
      const int l = tid >> 1, j = tid & 1;
      float s = 0.0f;
      for (int r = 0; r < 64; ++r) s += Ain[l * S68 + r] * vb[r * 2 + j];
      vb2[tid] = s;
    }
    __syncthreads();
    // s_raw = sum_j v_j^T Al v_j   (Aln = Al/amax folded in at the end)
    float part = 0.0f;
    for (int idx = tid; idx < 4096; idx += 256) {
      const int d = idx >> 6, u = idx & 63;
      part += Al[d * S68 + u] *
              (vb2[d * 2] * vb2[u * 2] + vb2[d * 2 + 1] * vb2[u * 2 + 1]);
    }
    const float sraw = block_reduce_sum(part, red, tid);
    if (tid == 0)
      out[b] = logsum + logf(amax) + logf(sraw / amax);
  }
}

// ---------------------------------------------------------------------------
// Kernel 2: scalar transfer-matrix chain (one workgroup).
//   M2' = sum_s W_s^T (M2/a) W_s  over 2 input / 4 output slices per site.
// ---------------------------------------------------------------------------
__global__ void __launch_bounds__(256)
norm_chain_kernel(const float* __restrict__ Mi,
                  const float* __restrict__ Mo,
                  float* __restrict__ out) {
  extern __shared__ float smem[];
  float* M2  = smem;
  float* M2n = M2 + MATF;
  float* W   = M2n + MATF;
  float* Tt  = W + MATF;
  float* red = Tt + MATF;
  const int tid = threadIdx.x;

  // init: Aln[k][l] = sum_ij Mo[0,0,k,i,j] * Mo[0,0,l,i,j]  (e00 seed folded)
  for (int idx = tid; idx < 4096; idx += 256) {
    const int kq = idx >> 6, l = idx & 63;
    const float* a4 = Mo + (size_t)kq * 4;
    const float* b4 = Mo + (size_t)l * 4;
    M2[kq * S68 + l] = a4[0] * b4[0] + a4[1] * b4[1] + a4[2] * b4[2] + a4[3] * b4[3];
  }
  prefetch_block((const char*)Mi, 4096 * 2 * 4, tid);
  prefetch_block((const char*)(Mo + (size_t)4096 * 4), 4096 * 4 * 4, tid);
  __syncthreads();

  float nlog = 0.0f;
  float* cur = M2;
  float* nxt = M2n;

  for (int k = 0; k < NIN; ++k) {
    // prefetch the next site's slabs; overlaps the 12 matmuls below
    if (k + 1 < NIN) {
      prefetch_block((const char*)(Mi + (size_t)(k + 1) * 8192), 4096 * 2 * 4, tid);
      prefetch_block((const char*)(Mo + (size_t)(k + 2) * 16384), 4096 * 4 * 4, tid);
    }
    // ---- normalize, then contract_in over 2 slices of Mi[k] ----
    {
      float lmax = 0.0f;
      for (int idx = tid; idx < 4096; idx += 256)
        lmax = fmaxf(lmax, fabsf(cur[(idx >> 6) * S68 + (idx & 63)]));
      const float a1 = block_reduce_max(lmax, red, tid);
      nlog += logf(a1);
      const float inv = 1.0f / a1;
      for (int idx = tid; idx < 4096; idx += 256)
        cur[(idx >> 6) * S68 + (idx & 63)] *= inv;
      __syncthreads();
      for (int i = 0; i < 2; ++i) {
        for (int idx = tid; idx < 4096; idx += 256)
          W[(idx >> 6) * S68 + (idx & 63)] = Mi[(size_t)k * 8192 + idx * 2 + i];
        __syncthreads();
        mm64<true>(W, cur, Tt, tid, false);   // T = W^T * cur
        __syncthreads();
        mm64<false>(Tt, W, nxt, tid, i > 0);  // nxt (+)= T * W
        __syncthreads();
      }
      float* t = cur; cur = nxt; nxt = t;
    }
    // ---- normalize, then contract_out over 4 slices of Mo[k+1] ----
    {
      float lmax = 0.0f;
      for (int idx = tid; idx < 4096; idx += 256)
        lmax = fmaxf(lmax, fabsf(cur[(idx >> 6) * S68 + (idx & 63)]));
      const float a2 = block_reduce_max(lmax, red, tid);
      nlog += logf(a2);
      const float inv = 1.0f / a2;
      for (int idx = tid; idx < 4096; idx += 256)
        cur[(idx >> 6) * S68 + (idx & 63)] *= inv;
      __syncthreads();
      for (int p = 0; p < 4; ++p) {
        for (int idx = tid; idx < 4096; idx += 256)
          W[(idx >> 6) * S68 + (idx & 63)] =
              Mo[(size_t)(k + 1) * 16384 + (size_t)idx * 4 + p];
        __syncthreads();
        mm64<true>(W, cur, Tt, tid, false);
        __syncthreads();
        mm64<false>(Tt, W, nxt, tid, p > 0);
        __syncthreads();
      }
      float* t = cur; cur = nxt; nxt = t;
    }
  }
  if (tid == 0) out[NB] = nlog + logf(cur[0]);
}

// ---------------------------------------------------------------------------
extern "C" void kernel_launch(void* const* d_in, const int* in_sizes, int n_in,
                              void* d_out, int out_size, void* d_ws, size_t ws_size,
                              hipStream_t stream) {
  (void)in_sizes; (void)n_in; (void)out_size; (void)d_ws; (void)ws_size;
  const float* x  = (const float*)d_in[0];   // (129,128,2)
  const float* Mi = (const float*)d_in[1];   // (64,64,64,2)
  const float* Mo = (const float*)d_in[2];   // (65,64,64,2,2)
  float* out = (float*)d_out;                // 129 floats: out[0..127], lnrm

  const size_t shm1 = (size_t)(7 * MATF + 256 + 128 + 128) * sizeof(float);
  const size_t shm2 = (size_t)(4 * MATF + 256) * sizeof(float);
  (void)hipFuncSetAttribute((const void*)batch_chain_kernel,
                            hipFuncAttributeMaxDynamicSharedMemorySize, (int)shm1);
  (void)hipFuncSetAttribute((const void*)norm_chain_kernel,
                            hipFuncAttributeMaxDynamicSharedMemorySize, (int)shm2);

  batch_chain_kernel<<<dim3(NB), dim3(256), shm1, stream>>>(x, Mi, Mo, out);
  norm_chain_kernel<<<dim3(1), dim3(256), shm2, stream>>>(Mi, Mo, out);
}